// CNN_1133871366670
// MI455X (gfx1250) — compile-verified
//
#include <hip/hip_runtime.h>
#include <hip/hip_bf16.h>

// Problem sizes (fixed by the reference)
#define BB   256   // batch
#define SS   64    // max sentence length
#define WWL  16    // max word length
#define DD   64    // char embedding dim
#define VV   512   // char vocab
#define KK   5     // conv taps
#define NTHREADS 1024   // SS*WWL, one thread per (word, 4-channel group)

typedef unsigned int u32x4 __attribute__((ext_vector_type(4)));
typedef int          i32x8 __attribute__((ext_vector_type(8)));
typedef int          i32x4 __attribute__((ext_vector_type(4)));

__device__ __forceinline__ float4 ld_tab(const float* __restrict__ tbl, int cid, int d0) {
    return *(const float4*)(tbl + cid * DD + d0);
}

__global__ __launch_bounds__(NTHREADS, 1)
void charcnn_kernel(const int*   __restrict__ char_ids,   // (B,S,W)
                    const int*   __restrict__ length,     // (B,)
                    const float* __restrict__ char_table, // (V,D)
                    const float* __restrict__ conv_w,     // (K,)
                    const float* __restrict__ conv_b,     // (1,)
                    float*       __restrict__ out)        // (B,S,D) then (B,) lengths
{
    __shared__ float tbl[VV * DD];       // 128 KB: whole char table, LDS-resident
    __shared__ int   cids[SS * WWL];     // 4 KB: this batch row's char ids

    const int tid = threadIdx.x;
    const int b   = blockIdx.x;

    // Stage char ids: exactly S*W = 1024 ints, one per thread.
    cids[tid] = char_ids[b * (SS * WWL) + tid];

    // ---- Stage the char table into LDS ----------------------------------
#if defined(__HIP_DEVICE_COMPILE__) && __has_builtin(__builtin_amdgcn_tensor_load_to_lds) && __has_builtin(__builtin_amdgcn_s_wait_tensorcnt)
    {
        // TDM path: each of the 32 waves DMAs its own 4 KB slice (16 table
        // rows = 1024 f32 elements) with a private descriptor. TDM ignores
        // EXEC and issues once per wave, so slicing per-wave avoids duplicate
        // transfers without any divergent-branch tricks.
        const unsigned wave  = (unsigned)tid >> 5;           // 0..31
        const unsigned elems = (VV * DD) / 32;               // 1024 f32
        typedef __attribute__((address_space(3))) float lds_f;
        const unsigned lds_base =
            (unsigned)(unsigned long long)(lds_f*)&tbl[0];
        const unsigned long long ga =
            (unsigned long long)(const void*)(char_table + wave * elems);
        const unsigned lds_addr = lds_base + wave * elems * 4u;

        // D# group 0: count=1, lds_addr, 57-bit global addr, type=2
        u32x4 g0;
        g0[0] = 1u;                                          // count=1, user mode
        g0[1] = lds_addr;                                    // lds_addr (bytes)
        g0[2] = (unsigned)(ga & 0xFFFFFFFFull);              // global_addr[31:0]
        g0[3] = (unsigned)((ga >> 32) & 0x01FFFFFFull)       // global_addr[56:32]
              | (2u << 30);                                  // type = 2 ("image")

        // D# group 1: wg_mask=0 (not in cluster), data_size=2 (4B),
        // 1-D tensor/tile of `elems` elements, stride = elems.
        i32x8 g1;
        g1[0] = (int)(2u << 16);                             // data_size = 4B
        g1[1] = (int)((elems & 0xFFFFu) << 16);              // tensor_dim0[15:0]
        g1[2] = (int)(((elems >> 16) & 0xFFFFu)              // tensor_dim0[31:16]
              | (1u << 16));                                 // tensor_dim1 = 1
        g1[3] = (int)((elems & 0xFFFFu) << 16);              // tile_dim0 = elems
        g1[4] = 0;                                           // tile_dim1/2 unused
        g1[5] = (int)elems;                                  // tensor_dim0_stride lo
        g1[6] = 0;                                           // stride hi / dim1_stride lo
        g1[7] = 0;                                           // dim1_stride hi
        i32x4 gz4 = {0, 0, 0, 0};                            // groups 2/3 unused (<=2D)
        i32x8 gz8 = {0, 0, 0, 0, 0, 0, 0, 0};                // extra group (clang-23 form)

        // clang-23 / therock-10.0 signature: 6 args
        __builtin_amdgcn_tensor_load_to_lds(g0, g1, gz4, gz4, gz8, 0);
        __builtin_amdgcn_s_wait_tensorcnt(0);                // own slice done
    }
#else
    {
        // Fallback: cooperative 128-bit vector copy (8 float4 per thread).
        const float4* src = (const float4*)char_table;
        float4*       dst = (float4*)tbl;
        #pragma unroll
        for (int i = 0; i < (VV * DD) / 4 / NTHREADS; ++i)
            dst[tid + i * NTHREADS] = src[tid + i * NTHREADS];
    }
#endif
    __syncthreads();   // all slices in LDS, cids staged

    // ---- Conv + max -----------------------------------------------------
    const int s  = tid >> 4;           // word index 0..63
    const int d0 = (tid & 15) << 2;    // channel base 0,4,...,60

    const int   len = length[b];       // uniform per block -> s_load
    const float cb  = conv_b[0];
    const float w0 = conv_w[0], w1 = conv_w[1], w2 = conv_w[2],
                w3 = conv_w[3], w4 = conv_w[4];

    float4 o;
    if (s >= len) {
        // Masked word: embedding is all-zero, conv = bias, max = bias.
        o.x = cb; o.y = cb; o.z = cb; o.w = cb;
    } else {
        const int* cw = &cids[s * WWL];
        const float4 zero = {0.f, 0.f, 0.f, 0.f};
        // Rolling window xpad[w-2..w+2]; 16 lanes of a wave hit the same
        // table row with consecutive float4s -> clean ds_load_b128s.
        float4 xa = zero, xb = zero;
        float4 xc = ld_tab(tbl, cw[0], d0);
        float4 xd = ld_tab(tbl, cw[1], d0);
        float4 xe = ld_tab(tbl, cw[2], d0);
        float4 m  = {-3.402823466e38f, -3.402823466e38f,
                     -3.402823466e38f, -3.402823466e38f};
        #pragma unroll
        for (int w = 0; w < WWL; ++w) {
            float4 c;
            c.x = cb + w0*xa.x + w1*xb.x + w2*xc.x + w3*xd.x + w4*xe.x;
            c.y = cb + w0*xa.y + w1*xb.y + w2*xc.y + w3*xd.y + w4*xe.y;
            c.z = cb + w0*xa.z + w1*xb.z + w2*xc.z + w3*xd.z + w4*xe.z;
            c.w = cb + w0*xa.w + w1*xb.w + w2*xc.w + w3*xd.w + w4*xe.w;
            m.x = fmaxf(m.x, c.x); m.y = fmaxf(m.y, c.y);
            m.z = fmaxf(m.z, c.z); m.w = fmaxf(m.w, c.w);
            xa = xb; xb = xc; xc = xd; xd = xe;
            xe = (w + 3 < WWL) ? ld_tab(tbl, cw[w + 3], d0) : zero;
        }
        o = m;
    }
    *(float4*)(out + b * (SS * DD) + s * DD + d0) = o;
}

// Tail: second tuple output `length`, converted to the float output dtype
// (values <= 64, exactly representable).
__global__ void length_tail_kernel(const int* __restrict__ length,
                                   float* __restrict__ out)
{
    const int i = threadIdx.x;
    if (i < BB) out[BB * SS * DD + i] = (float)length[i];
}

extern "C" void kernel_launch(void* const* d_in, const int* in_sizes, int n_in,
                              void* d_out, int out_size, void* d_ws, size_t ws_size,
                              hipStream_t stream) {
    const int*   char_ids   = (const int*)d_in[0];
    const int*   length     = (const int*)d_in[1];
    const float* char_table = (const float*)d_in[2];
    const float* conv_w     = (const float*)d_in[3];
    const float* conv_b     = (const float*)d_in[4];
    float* out = (float*)d_out;

    charcnn_kernel<<<BB, NTHREADS, 0, stream>>>(char_ids, length, char_table,
                                                conv_w, conv_b, out);
    length_tail_kernel<<<1, BB, 0, stream>>>(length, out);
}